// se3_Encoder_26645977105058
// MI455X (gfx1250) — compile-verified
//
#include <hip/hip_runtime.h>
#include <math.h>

#define NN   16384
#define CC   256
#define HH   8
#define DD   64
#define KK   8
#define HID  512
#define OUTD 64
#define NE   64
#define RHID 16

typedef float v2f __attribute__((ext_vector_type(2)));
typedef float v8f __attribute__((ext_vector_type(8)));
typedef unsigned int u32x4 __attribute__((ext_vector_type(4)));
typedef int i32x4 __attribute__((ext_vector_type(4)));
typedef int i32x8 __attribute__((ext_vector_type(8)));

__device__ __forceinline__ float gelu_f(float x) {
    // jax.nn.gelu default (approximate=True, tanh form)
    const float c0 = 0.7978845608028654f;
    float x3 = x * x * x;
    return 0.5f * x * (1.0f + tanhf(c0 * (x + 0.044715f * x3)));
}

// ---------------------------------------------------------------------------
// fp32 WMMA GEMM with TDM-staged A tile.
// One block = 4 waves = one 16-row M-tile x four 16-col N-tiles.
// Wave 0 issues a single tensor_load_to_lds of the 16 x Kdim A-strip
// (2D descriptor: tile_dim0 = Kdim, tile_dim1 = 16, stride = lda), waits on
// TENSORcnt, then all waves consume A from LDS while B streams from global.
// EPI: 0 = none, 1 = tanh(a1*(res+acc))*g+b, 2 = tanh(a2*gelu(acc+bias))*g+b,
//      3 = tanh(gelu(acc+bias))
// ---------------------------------------------------------------------------
template <int EPI>
__global__ __launch_bounds__(128) void gemm_wmma_f32(
    const float* __restrict__ A, int lda,
    const float* __restrict__ B, int ldb,
    float* __restrict__ C, int ldc,
    int M, int Ncols, int Kdim,
    const float* __restrict__ bias,
    const float* __restrict__ res, int ldres,
    const float* __restrict__ alpha,
    const float* __restrict__ gamma,
    const float* __restrict__ beta)
{
    extern __shared__ float ldsA[];   // [16][Kdim] row-major, contiguous

    const int wave = threadIdx.x >> 5;
    const int lane = threadIdx.x & 31;
    const int groupsN = Ncols >> 6;               // N-tile groups of 4
    const int tm = blockIdx.x / groupsN;
    const int tg = blockIdx.x - tm * groupsN;
    const int r0 = tm << 4;
    const int n0 = ((tg << 2) + wave) << 4;
    const int m  = lane & 15;          // row (A) / col (B,C) within tile
    const int kq = (lane >> 4) << 1;   // 0 or 2: K sub-offset for this half-wave

    // ---- TDM: DMA the 16 x Kdim A-strip into LDS (wave 0 only) ----
    if (threadIdx.x < 32) {
        unsigned long long ga = (unsigned long long)(const void*)(A + (size_t)r0 * lda);
        unsigned int lds_off = (unsigned int)(size_t)(&ldsA[0]);  // LDS aperture: addr[31:0]
        u32x4 g0;
        g0[0] = 1u;                                   // count=1, is_restore=0
        g0[1] = lds_off;                              // lds_addr
        g0[2] = (unsigned int)ga;                     // global_addr[31:0]
        g0[3] = ((unsigned int)(ga >> 32) & 0x01FFFFFFu) | (2u << 30);  // addr[56:32] | type=2
        i32x8 g1;
        g1[0] = (int)(2u << 16);                      // data_size=2 (4B), no pad/iter/mask
        g1[1] = (int)(((unsigned)Kdim & 0xFFFFu) << 16);            // tensor_dim0[15:0]
        g1[2] = (int)((((unsigned)Kdim >> 16) & 0xFFFFu)
                      | (((unsigned)M & 0xFFFFu) << 16));           // dim0 hi | tensor_dim1 lo
        g1[3] = (int)((((unsigned)M >> 16) & 0xFFFFu)
                      | (((unsigned)Kdim & 0xFFFFu) << 16));        // dim1 hi | tile_dim0=Kdim
        g1[4] = 16;                                   // tile_dim1=16, tile_dim2=0
        g1[5] = (int)(unsigned)lda;                   // tensor_dim0_stride[31:0]
        g1[6] = 0;                                    // stride hi | dim1_stride lo
        g1[7] = 0;
        i32x4 z4 = {0, 0, 0, 0};
        i32x8 z8 = {0, 0, 0, 0, 0, 0, 0, 0};
        __builtin_amdgcn_tensor_load_to_lds(g0, g1, z4, z4, z8, 0);
        __builtin_amdgcn_s_wait_tensorcnt(0);
    }
    __syncthreads();

    // ---- WMMA K-loop: A from LDS, B from global ----
    v8f acc = {};
    const int aoff = m * Kdim + kq;
    const float* Bcol = B + n0 + m;

    for (int kb = 0; kb < Kdim; kb += 4) {
        v2f a, b;
        a.x = ldsA[aoff + kb];
        a.y = ldsA[aoff + kb + 1];
        b.x = Bcol[(size_t)(kb + kq) * ldb];
        b.y = Bcol[(size_t)(kb + kq + 1) * ldb];
        acc = __builtin_amdgcn_wmma_f32_16x16x4_f32(
            false, a, false, b, (short)0, acc, false, false);
    }

    const int col = n0 + m;
    float a_v = 0.0f, g_v = 0.0f, b_v = 0.0f, bias_v = 0.0f;
    if (EPI == 1 || EPI == 2) { a_v = alpha[0]; g_v = gamma[col]; b_v = beta[col]; }
    if (EPI == 2 || EPI == 3) { bias_v = bias[col]; }

    const int rbase = r0 + ((lane >> 4) << 3);
#pragma unroll
    for (int r = 0; r < 8; ++r) {
        const int row = rbase + r;
        float v = acc[r];
        if (EPI == 1) {
            v += res[(size_t)row * ldres + col];
            v = tanhf(a_v * v) * g_v + b_v;
        } else if (EPI == 2) {
            v = gelu_f(v + bias_v);
            v = tanhf(a_v * v) * g_v + b_v;
        } else if (EPI == 3) {
            v = tanhf(gelu_f(v + bias_v));
        }
        C[(size_t)row * ldc + col] = v;
    }
}

// ---------------------------------------------------------------------------
// Gate projection: G[n,h] = x[n,:] . w_gate[:,h]   (Nc = 8, too thin for WMMA)
// ---------------------------------------------------------------------------
__global__ __launch_bounds__(256) void gate_kernel(
    const float* __restrict__ x, const float* __restrict__ wg,
    float* __restrict__ G)
{
    int id = blockIdx.x * blockDim.x + threadIdx.x;
    if (id >= NN * HH) return;
    int n = id >> 3, h = id & 7;
    const float* xr = x + (size_t)n * CC;
    float s = 0.0f;
    for (int c = 0; c < CC; ++c) s += xr[c] * wg[c * HH + h];
    G[id] = s;
}

// ---------------------------------------------------------------------------
// SE(3) sparse-neighbor attention. 64 threads per node, 4 nodes per block.
// Writes AO[n, 0:512] = out0 and AO[n, 512+h] = vecnorm[h].
// ---------------------------------------------------------------------------
__global__ __launch_bounds__(256) void attn_kernel(
    const float* __restrict__ Q, const float* __restrict__ KM,
    const float* __restrict__ VM, const float* __restrict__ G,
    const float* __restrict__ coors, const int* __restrict__ nbr,
    const float* __restrict__ Wr1, const float* __restrict__ br1,
    const float* __restrict__ Wr2, const float* __restrict__ br2,
    float* __restrict__ AO)
{
    __shared__ int   s_nb[4][KK];
    __shared__ float s_dist[4][KK];
    __shared__ float s_unit[4][KK][3];
    __shared__ float s_logit[4][KK][HH];
    __shared__ float s_attn[4][KK][HH];

    const int slot = threadIdx.y;
    const int node = blockIdx.x * 4 + slot;
    const int t = threadIdx.x;   // 0..63

    if (t < KK) {
        int nb = nbr[(size_t)node * KK + t];
        s_nb[slot][t] = nb;
        float rx = coors[(size_t)nb * 3 + 0] - coors[(size_t)node * 3 + 0];
        float ry = coors[(size_t)nb * 3 + 1] - coors[(size_t)node * 3 + 1];
        float rz = coors[(size_t)nb * 3 + 2] - coors[(size_t)node * 3 + 2];
        float d = sqrtf(rx * rx + ry * ry + rz * rz + 1e-8f);
        s_dist[slot][t] = d;
        float inv = 1.0f / d;
        s_unit[slot][t][0] = rx * inv;
        s_unit[slot][t][1] = ry * inv;
        s_unit[slot][t][2] = rz * inv;
    }
    __syncthreads();

    {   // logits: one (k,h) pair per thread
        const int k = t >> 3, h = t & 7;
        const float dist = s_dist[slot][k];
        float rad = br2[h];
        for (int j = 0; j < RHID; ++j)
            rad += gelu_f(dist * Wr1[j] + br1[j]) * Wr2[j * HH + h];
        const int nb = s_nb[slot][k];
        const float* qp = Q + (size_t)node * (HH * DD) + h * DD;
        const float* kp = KM + (size_t)nb * (HH * DD) + h * DD;
        float dotv = 0.0f;
        for (int d = 0; d < DD; ++d) dotv += qp[d] * kp[d];
        float lg = dotv * 0.125f + rad;            // 1/sqrt(64)
        if (dist > 10.0f) lg = -1e9f;
        s_logit[slot][k][h] = lg;
    }
    __syncthreads();

    if (t < HH) {   // softmax over K, degree-1 vector + norm
        const int h = t;
        float mx = -3.4e38f;
        for (int k = 0; k < KK; ++k) mx = fmaxf(mx, s_logit[slot][k][h]);
        float e[KK], sum = 0.0f;
        for (int k = 0; k < KK; ++k) { e[k] = expf(s_logit[slot][k][h] - mx); sum += e[k]; }
        float inv = 1.0f / sum;
        float vx = 0.0f, vy = 0.0f, vz = 0.0f;
        for (int k = 0; k < KK; ++k) {
            float a = e[k] * inv;
            s_attn[slot][k][h] = a;
            float g = G[(size_t)s_nb[slot][k] * HH + h];
            float ag = a * g;
            vx += ag * s_unit[slot][k][0];
            vy += ag * s_unit[slot][k][1];
            vz += ag * s_unit[slot][k][2];
        }
        AO[(size_t)node * 520 + 512 + h] = sqrtf(vx * vx + vy * vy + vz * vz + 1e-8f);
    }
    __syncthreads();

    {   // out0: each thread handles 8 of the 512 outputs
        const int h = t >> 3;
        const int db = (t & 7) << 3;
        float o[8] = {0, 0, 0, 0, 0, 0, 0, 0};
        for (int k = 0; k < KK; ++k) {
            float a = s_attn[slot][k][h];
            const float* vp = VM + (size_t)s_nb[slot][k] * (HH * DD) + h * DD + db;
#pragma unroll
            for (int d = 0; d < 8; ++d) o[d] += a * vp[d];
        }
        float* op = AO + (size_t)node * 520 + h * DD + db;
#pragma unroll
        for (int d = 0; d < 8; ++d) op[d] = o[d];
    }
}

// ---------------------------------------------------------------------------
// Concat tail: H2[n, 512+j] = tanh(alpha2 * x[n,j]) * gamma2[512+j] + beta2[512+j]
// ---------------------------------------------------------------------------
__global__ __launch_bounds__(256) void concat_dt2_kernel(
    const float* __restrict__ x, const float* __restrict__ alpha2,
    const float* __restrict__ gamma2, const float* __restrict__ beta2,
    float* __restrict__ H2)
{
    int id = blockIdx.x * blockDim.x + threadIdx.x;
    if (id >= NN * 20) return;
    int n = id / 20, j = id - n * 20;
    float v = tanhf(alpha2[0] * x[(size_t)n * CC + j]) * gamma2[512 + j] + beta2[512 + j];
    H2[(size_t)n * 532 + 512 + j] = v;
}

// ---------------------------------------------------------------------------
// VQ: nearest codebook row (first-index tie break), straight-through output,
// per-node squared-error partial into lossbuf[node] (deterministic).
// ---------------------------------------------------------------------------
__global__ __launch_bounds__(256) void vq_kernel(
    const float* __restrict__ Z, const float* __restrict__ cb,
    float* __restrict__ out, float* __restrict__ lossbuf)
{
    __shared__ float s_z[4][NE];
    __shared__ float s_d2[4][NE];
    __shared__ int   s_i[4][NE];
    __shared__ float s_s[4][NE];

    const int slot = threadIdx.y;
    const int node = blockIdx.x * 4 + slot;
    const int t = threadIdx.x;   // 0..63

    s_z[slot][t] = Z[(size_t)node * OUTD + t];
    __syncthreads();

    const float* ce = cb + (size_t)t * OUTD;
    float d2 = 0.0f;
    for (int d = 0; d < OUTD; ++d) {
        float df = s_z[slot][d] - ce[d];
        d2 += df * df;
    }
    s_d2[slot][t] = d2;
    s_i[slot][t] = t;
    __syncthreads();

    for (int off = 32; off > 0; off >>= 1) {
        if (t < off) {
            float v1 = s_d2[slot][t], v2 = s_d2[slot][t + off];
            int i1 = s_i[slot][t], i2 = s_i[slot][t + off];
            if (v2 < v1 || (v2 == v1 && i2 < i1)) {
                s_d2[slot][t] = v2;
                s_i[slot][t] = i2;
            }
        }
        __syncthreads();
    }
    const int best = s_i[slot][0];

    float zq = cb[(size_t)best * OUTD + t];
    out[(size_t)node * OUTD + t] = zq;   // z + sg(zq - z) == zq numerically
    float df = s_z[slot][t] - zq;
    s_s[slot][t] = df * df;
    __syncthreads();
    for (int off = 32; off > 0; off >>= 1) {
        if (t < off) s_s[slot][t] += s_s[slot][t + off];
        __syncthreads();
    }
    if (t == 0) lossbuf[node] = s_s[slot][0];
}

__global__ __launch_bounds__(256) void loss_finalize_kernel(
    const float* __restrict__ lossbuf, float* __restrict__ out)
{
    __shared__ float s[256];
    float acc = 0.0f;
    for (int i = threadIdx.x; i < NN; i += 256) acc += lossbuf[i];
    s[threadIdx.x] = acc;
    __syncthreads();
    for (int off = 128; off > 0; off >>= 1) {
        if (threadIdx.x < off) s[threadIdx.x] += s[threadIdx.x + off];
        __syncthreads();
    }
    if (threadIdx.x == 0)
        out[(size_t)NN * OUTD] = 0.25f * s[0] / (float)((size_t)NN * OUTD);
}

// ---------------------------------------------------------------------------
extern "C" void kernel_launch(void* const* d_in, const int* in_sizes, int n_in,
                              void* d_out, int out_size, void* d_ws, size_t ws_size,
                              hipStream_t stream)
{
    (void)in_sizes; (void)n_in; (void)out_size; (void)ws_size;
    const float* x      = (const float*)d_in[0];
    const float* coors  = (const float*)d_in[1];
    const int*   nbr    = (const int*)d_in[2];
    const float* Wq     = (const float*)d_in[3];
    const float* Wk     = (const float*)d_in[4];
    const float* Wv     = (const float*)d_in[5];
    const float* wg     = (const float*)d_in[6];
    const float* Wr1    = (const float*)d_in[7];
    const float* br1    = (const float*)d_in[8];
    const float* Wr2    = (const float*)d_in[9];
    const float* br2    = (const float*)d_in[10];
    const float* Wo     = (const float*)d_in[11];
    const float* alpha1 = (const float*)d_in[12];
    const float* gamma1 = (const float*)d_in[13];
    const float* beta1  = (const float*)d_in[14];
    const float* Wlin   = (const float*)d_in[15];
    const float* blin   = (const float*)d_in[16];
    const float* alpha2 = (const float*)d_in[17];
    const float* gamma2 = (const float*)d_in[18];
    const float* beta2  = (const float*)d_in[19];
    const float* Wout   = (const float*)d_in[20];
    const float* bout   = (const float*)d_in[21];
    const float* cb     = (const float*)d_in[22];
    float* out = (float*)d_out;

    float* ws = (float*)d_ws;
    float* Q  = ws;                          // [N,512]
    float* KM = ws + (size_t)NN * 512;       // [N,512]
    float* VM = ws + (size_t)NN * 1024;      // [N,512]
    float* G  = ws + (size_t)NN * 1536;      // [N,8]
    float* AO = ws + (size_t)NN * 1544;      // [N,520]
    float* Z  = ws + (size_t)NN * 2064;      // [N,64]
    float* LB = ws + (size_t)NN * 2128;      // [N] per-node loss partials
    float* T1 = Q;                           // [N,256]  (Q dead after attention)
    float* H2 = KM;                          // [N,532]  (KM/VM dead after attention)

    // 1-3: Q/K/V projections, 16384x512 <- 16384x256 @ 256x512
    {
        dim3 g((NN / 16) * (512 / 64)), b(128);
        size_t sh = (size_t)CC * 16 * sizeof(float);
        gemm_wmma_f32<0><<<g, b, sh, stream>>>(x, CC, Wq, 512, Q, 512, NN, 512, CC,
                                               nullptr, nullptr, 0, nullptr, nullptr, nullptr);
        gemm_wmma_f32<0><<<g, b, sh, stream>>>(x, CC, Wk, 512, KM, 512, NN, 512, CC,
                                               nullptr, nullptr, 0, nullptr, nullptr, nullptr);
        gemm_wmma_f32<0><<<g, b, sh, stream>>>(x, CC, Wv, 512, VM, 512, NN, 512, CC,
                                               nullptr, nullptr, 0, nullptr, nullptr, nullptr);
    }
    // 4: gate projection (8 cols)
    gate_kernel<<<dim3((NN * HH + 255) / 256), dim3(256), 0, stream>>>(x, wg, G);

    // 5: attention -> AO [N,520]
    attn_kernel<<<dim3(NN / 4), dim3(64, 4), 0, stream>>>(
        Q, KM, VM, G, coors, nbr, Wr1, br1, Wr2, br2, AO);

    // 6: T1 = tanh(alpha1*(x + AO@Wo))*gamma1+beta1, 16384x256 <- 16384x520 @ 520x256
    {
        dim3 g((NN / 16) * (CC / 64)), b(128);
        size_t sh = (size_t)520 * 16 * sizeof(float);
        gemm_wmma_f32<1><<<g, b, sh, stream>>>(AO, 520, Wo, CC, T1, CC, NN, CC, 520,
                                               nullptr, x, CC, alpha1, gamma1, beta1);
    }
    // 7: H2[:, :512] = tanh(alpha2*gelu(T1@Wlin+blin))*gamma2+beta2 (ldc=532)
    {
        dim3 g((NN / 16) * (HID / 64)), b(128);
        size_t sh = (size_t)CC * 16 * sizeof(float);
        gemm_wmma_f32<2><<<g, b, sh, stream>>>(T1, CC, Wlin, HID, H2, 532, NN, HID, CC,
                                               blin, nullptr, 0, alpha2, gamma2, beta2);
    }
    // 7b: H2[:, 512:532] from x[:, :20]
    concat_dt2_kernel<<<dim3((NN * 20 + 255) / 256), dim3(256), 0, stream>>>(
        x, alpha2, gamma2, beta2, H2);

    // 8: Z = tanh(gelu(H2@Wout + bout)), 16384x64 <- 16384x532 @ 532x64
    {
        dim3 g((NN / 16) * (OUTD / 64)), b(128);
        size_t sh = (size_t)532 * 16 * sizeof(float);
        gemm_wmma_f32<3><<<g, b, sh, stream>>>(H2, 532, Wout, OUTD, Z, OUTD, NN, OUTD, 532,
                                               bout, nullptr, 0, nullptr, nullptr, nullptr);
    }
    // 9: VQ + deterministic loss reduction
    vq_kernel<<<dim3(NN / 4), dim3(64, 4), 0, stream>>>(Z, cb, out, LB);
    loss_finalize_kernel<<<dim3(1), dim3(256), 0, stream>>>(LB, out);
}